// HashEncoder4D_86268713107703
// MI455X (gfx1250) — compile-verified
//
#include <hip/hip_runtime.h>
#include <stdint.h>

#define N_LEVELS      16
#define LOG2_HASH     15
#define HASHMAP_SIZE  (1 << LOG2_HASH)       // 32768 entries
#define FEAT_DIM      2
#define TABLE_BYTES   (HASHMAP_SIZE * FEAT_DIM * 4)   // 262144 B = 256 KB per level
#define BLOCK_THREADS 512
#define BLOCKS_PER_LEVEL 64

// Resolutions from exp(lerp(log 16, log 512)) over 16 levels (Instant-NGP schedule).
__constant__ float c_res[N_LEVELS] = {
    16.f, 20.f, 25.f, 32.f, 40.f, 50.f, 64.f, 80.f,
    101.f, 128.f, 161.f, 203.f, 256.f, 322.f, 406.f, 512.f
};

extern "C" __global__ __launch_bounds__(BLOCK_THREADS, 1)
void hashenc4d_kernel(const float* __restrict__ coords,
                      const float* __restrict__ timestamps,
                      const float* __restrict__ tables,
                      float* __restrict__ out,
                      int n_points)
{
    extern __shared__ float2 s_tab[];          // 256 KB: this level's full hash table
    const int level = blockIdx.y;

    // ------------------------------------------------------------------
    // Stage this level's 256 KB table into LDS via CDNA5 async loads
    // (tracked by ASYNCcnt). GVS form: addr = SGPR64 base + VGPR offset.
    // The VDST VGPR carries the workgroup-relative LDS byte address.
    // ------------------------------------------------------------------
    {
        const char* gtab = (const char*)tables + (size_t)level * TABLE_BYTES;
        uint32_t lds_base = (uint32_t)(uintptr_t)(&s_tab[0]);
        for (int i = threadIdx.x; i < TABLE_BYTES / 16; i += BLOCK_THREADS) {
            uint32_t goff = (uint32_t)i * 16u;
            uint32_t lds_addr = lds_base + goff;
            asm volatile("global_load_async_to_lds_b128 %0, %1, %2"
                         :
                         : "v"(lds_addr), "v"(goff), "s"(gtab)
                         : "memory");
        }
        asm volatile("s_wait_asynccnt 0x0" ::: "memory");
        __syncthreads();
    }

    const float res = c_res[level];

    const int tid    = blockIdx.x * BLOCK_THREADS + threadIdx.x;
    const int stride = gridDim.x * BLOCK_THREADS;

    for (int n = tid; n < n_points; n += stride) {
        // normalize: xyz -> (c + 50) / 100 ; t -> clip(t, 0, 1)
        float c0 = (coords[3 * n + 0] + 50.0f) / 100.0f;
        float c1 = (coords[3 * n + 1] + 50.0f) / 100.0f;
        float c2 = (coords[3 * n + 2] + 50.0f) / 100.0f;
        float t  = timestamps[n];
        t = t < 0.0f ? 0.0f : (t > 1.0f ? 1.0f : t);
        float c[4] = { c0, c1, c2, t };

        // Per-dimension hashed corner terms and interpolation weights.
        const uint32_t pr[4] = { 1u, 2654435761u, 805459861u, 3674653429u };
        uint32_t h0[4], h1[4];
        float    w0[4], w1[4];
        #pragma unroll
        for (int d = 0; d < 4; ++d) {
            float s  = c[d] * res;
            float g  = floorf(s);
            float fr = s - g;
            int   gi = (int)g;
            h0[d] = (uint32_t)gi * pr[d];
            h1[d] = ((uint32_t)gi + 1u) * pr[d];
            w0[d] = 1.0f - fr;
            w1[d] = fr;
        }

        // Factor the 4D corner enumeration into two 2D halves:
        // corner ci = (b3 b2 b1 b0); hash = h01[ci&3] ^ h23[ci>>2],
        // weight = w01[ci&3] * w23[ci>>2]. Saves ~1/3 of corner-loop VALU.
        uint32_t h01[4], h23[4];
        float    w01[4], w23[4];
        #pragma unroll
        for (int j = 0; j < 4; ++j) {
            h01[j] = ((j & 1) ? h1[0] : h0[0]) ^ ((j & 2) ? h1[1] : h0[1]);
            w01[j] = ((j & 1) ? w1[0] : w0[0]) * ((j & 2) ? w1[1] : w0[1]);
            h23[j] = ((j & 1) ? h1[2] : h0[2]) ^ ((j & 2) ? h1[3] : h0[3]);
            w23[j] = ((j & 1) ? w1[2] : w0[2]) * ((j & 2) ? w1[3] : w0[3]);
        }

        float acc0 = 0.0f, acc1 = 0.0f;
        #pragma unroll
        for (int ci = 0; ci < 16; ++ci) {
            uint32_t h = h01[ci & 3] ^ h23[ci >> 2];
            float    w = w01[ci & 3] * w23[ci >> 2];
            float2  f2 = s_tab[h & (uint32_t)(HASHMAP_SIZE - 1)];  // ds_load_b64
            acc0 = fmaf(w, f2.x, acc0);
            acc1 = fmaf(w, f2.y, acc1);
        }

        // out[n, level*2 + {0,1}] ; row = 32 floats, 8B-aligned float2 store
        float2* op = (float2*)(out + (size_t)n * (2 * N_LEVELS) + 2 * level);
        *op = make_float2(acc0, acc1);
    }
}

extern "C" void kernel_launch(void* const* d_in, const int* in_sizes, int n_in,
                              void* d_out, int out_size, void* d_ws, size_t ws_size,
                              hipStream_t stream)
{
    const float* coords     = (const float*)d_in[0];   // [N, 3]
    const float* timestamps = (const float*)d_in[1];   // [N, 1]
    const float* tables     = (const float*)d_in[2];   // [16, 32768, 2]
    float*       out        = (float*)d_out;           // [N, 32]
    int n_points = in_sizes[0] / 3;

    (void)hipFuncSetAttribute((const void*)hashenc4d_kernel,
                              hipFuncAttributeMaxDynamicSharedMemorySize,
                              TABLE_BYTES);

    dim3 grid(BLOCKS_PER_LEVEL, N_LEVELS, 1);
    dim3 block(BLOCK_THREADS, 1, 1);
    hashenc4d_kernel<<<grid, block, TABLE_BYTES, stream>>>(
        coords, timestamps, tables, out, n_points);
}